// GMMLayer_47201690583088
// MI455X (gfx1250) — compile-verified
//
#include <hip/hip_runtime.h>
#include <hip/hip_bf16.h>

// ---------------------------------------------------------------------------
// GMMConv (MoNet) layer for MI455X / gfx1250.
// N=50000 nodes, E=1.6M edges, IN=OUT=128, K=3 kernels, D=2.
// GEMM: V_WMMA_F32_16X16X4_F32 with double-buffered async global->LDS staging
// (GLOBAL_LOAD_ASYNC_TO_LDS_B128 / s_wait_asynccnt — CDNA5 async path).
// Edge aggregation: b128 gathers + fused K-blend + global_atomic_add_f32.
// ---------------------------------------------------------------------------

typedef __attribute__((ext_vector_type(2))) float v2f;
typedef __attribute__((ext_vector_type(8))) float v8f;

#define IN_DIM   128
#define OUT_DIM  128
#define KKER     3
#define KOUT     384   // KKER * OUT_DIM
#define BN_EPS   1e-5f

__device__ __forceinline__ void atomAddF32(float* p, float v) {
  unsafeAtomicAdd(p, v);   // -> global_atomic_add_f32
}

// Async copy of 16 bytes: cache/memory -> LDS, no VGPR staging (ASYNCcnt).
// Generic LDS pointer's low 32 bits are the wave-relative LDS byte offset.
__device__ __forceinline__ void async_copy16(const void* gptr, void* lptr) {
  unsigned lds = (unsigned)(size_t)lptr;
  asm volatile("global_load_async_to_lds_b128 %0, %1, off"
               :: "v"(lds), "v"(gptr) : "memory");
}
__device__ __forceinline__ void wait_async0() {
  asm volatile("s_wait_asynccnt 0x0" ::: "memory");
}

// ---------------------------------------------------------------------------
// Kernel 0: zero a float region (agg + deg + BN accumulators).
// ---------------------------------------------------------------------------
__global__ __launch_bounds__(256) void zero_f32(float* __restrict__ p, long long n) {
  long long i = (long long)blockIdx.x * blockDim.x + threadIdx.x;
  long long stride = (long long)gridDim.x * blockDim.x;
  for (; i < n; i += stride) p[i] = 0.0f;
}

// ---------------------------------------------------------------------------
// Kernel 1: xp[N,384] = feature[N,128] @ weight[128,384] via f32 WMMA.
// Block = 256 threads = 8 waves; block tile 64(M) x 64(N); K pipelined in
// four 32-deep chunks through a 2-deep async-LDS double buffer.
// A-frag (ISA 7.12.2, 32-bit A 16x4): lanes 0-15 -> K+0/K+1 in v0/v1,
// lanes 16-31 -> K+2/K+3. B-frag (4x16): v0 -> K+0 (lanes 0-15) / K+2
// (lanes 16-31); v1 -> K+1 / K+3. C/D: VGPR r -> rows r and r+8.
// ---------------------------------------------------------------------------
__global__ __launch_bounds__(256) void gemm_xp(const float* __restrict__ feature,
                                               const float* __restrict__ weight,
                                               float* __restrict__ xp,
                                               int nNodes) {
  // Padded for conflict-free fragment reads:
  // As stride 36 dwords -> banks k + 4*(9m mod 16), all 32 lanes distinct.
  // Bs stride 72 dwords -> half-lanes hit disjoint 16-bank ranges.
  __shared__ float As[2][64 * 36];   // 2 x 9.0 KB
  __shared__ float Bs[2][32 * 72];   // 2 x 9.0 KB

  const int t    = threadIdx.x;
  const int lane = t & 31;
  const int wave = t >> 5;
  const int half = lane >> 4;   // 0: lanes 0-15, 1: lanes 16-31
  const int l16  = lane & 15;
  const int rowBase = blockIdx.y * 64;
  const int colBase = blockIdx.x * 64;
  const int tM = (wave & 3) * 16;       // wave's M sub-tile
  const int n0 = (wave >> 2) * 32;      // wave's N strip (two 16-wide tiles)

  v8f c0 = {};
  v8f c1 = {};

  // Issue one K-chunk (64 rows x 32 k of A, 32 k x 64 cols of B) as async
  // b128 copies. Out-of-range rows are clamped to a valid source row: A-row m
  // only feeds D-row m, and out-of-range D rows are masked at the store.
  auto stage = [&](int ks, int buf) {
#pragma unroll
    for (int j = 0; j < 2; ++j) {
      int i  = t + j * 256;        // float4 index: 8 per A row
      int r  = i >> 3;
      int cc = (i & 7) * 4;
      int gr = rowBase + r;
      gr = gr < nNodes ? gr : (nNodes - 1);
      async_copy16(feature + (size_t)gr * IN_DIM + ks + cc, &As[buf][r * 36 + cc]);
    }
#pragma unroll
    for (int j = 0; j < 2; ++j) {
      int i  = t + j * 256;        // float4 index: 16 per B row
      int r  = i >> 4;
      int cc = (i & 15) * 4;
      async_copy16(weight + (size_t)(ks + r) * KOUT + colBase + cc, &Bs[buf][r * 72 + cc]);
    }
  };

  auto compute = [&](int buf) {
#pragma unroll
    for (int kk = 0; kk < 32; kk += 4) {
      const int ka = kk + 2 * half;
      v2f a, b0, b1;
      a.x  = As[buf][(tM + l16) * 36 + ka];
      a.y  = As[buf][(tM + l16) * 36 + ka + 1];
      b0.x = Bs[buf][(ka)     * 72 + n0 + l16];
      b0.y = Bs[buf][(ka + 1) * 72 + n0 + l16];
      b1.x = Bs[buf][(ka)     * 72 + n0 + 16 + l16];
      b1.y = Bs[buf][(ka + 1) * 72 + n0 + 16 + l16];
      c0 = __builtin_amdgcn_wmma_f32_16x16x4_f32(false, a, false, b0, (short)0, c0, false, false);
      c1 = __builtin_amdgcn_wmma_f32_16x16x4_f32(false, a, false, b1, (short)0, c1, false, false);
    }
  };

  // Software pipeline: async fill of chunk s+1 overlaps WMMA on chunk s.
  stage(0, 0);
  wait_async0();
  __syncthreads();
#pragma unroll
  for (int s = 0; s < 4; ++s) {
    if (s + 1 < 4) stage((s + 1) * 32, (s + 1) & 1);
    compute(s & 1);
    if (s + 1 < 4) {
      wait_async0();      // next chunk's copies have landed (own wave)
      __syncthreads();    // ...for every wave; also fences buffer reuse
    }
  }

  // Store C: VGPR r holds rows (r, r+8); column = lane&15 within the tile.
#pragma unroll
  for (int r = 0; r < 8; ++r) {
    int row = rowBase + tM + r + 8 * half;
    if (row < nNodes) {
      xp[(size_t)row * KOUT + colBase + n0 + l16]      = c0[r];
      xp[(size_t)row * KOUT + colBase + n0 + 16 + l16] = c1[r];
    }
  }
}

// ---------------------------------------------------------------------------
// Kernel 2: edge aggregation. 32 lanes per edge, 8 edges per 256-thread block.
// Each lane handles 4 channels (float4). Gaussian weights computed inline,
// K-blend fused before the scatter so each channel costs ONE atomic add.
// ---------------------------------------------------------------------------
__global__ __launch_bounds__(256) void edge_agg(const float* __restrict__ xp,
                                                const float* __restrict__ edge_attr,
                                                const int* __restrict__ eidx,
                                                const float* __restrict__ mu,
                                                const float* __restrict__ isig,
                                                float* __restrict__ agg,
                                                float* __restrict__ deg,
                                                int nEdges) {
  const int lane = threadIdx.x & 31;
  const int e = blockIdx.x * 8 + (threadIdx.x >> 5);
  if (e >= nEdges) return;

  const int src = eidx[e];            // edge_index[0][e]
  const int dst = eidx[nEdges + e];   // edge_index[1][e]

  const float u0 = edge_attr[2 * e];
  const float u1 = edge_attr[2 * e + 1];
  float w[KKER];
#pragma unroll
  for (int k = 0; k < KKER; ++k) {
    float d0 = (u0 - mu[2 * k])     * isig[2 * k];
    float d1 = (u1 - mu[2 * k + 1]) * isig[2 * k + 1];
    w[k] = __expf(-0.5f * (d0 * d0 + d1 * d1));
  }

  const int c = lane * 4;
  const float* base = xp + (size_t)src * KOUT + c;
  float4 x0 = *(const float4*)(base);             // kernel 0
  float4 x1 = *(const float4*)(base + OUT_DIM);   // kernel 1
  float4 x2 = *(const float4*)(base + 2 * OUT_DIM);

  float4 m;
  m.x = w[0] * x0.x + w[1] * x1.x + w[2] * x2.x;
  m.y = w[0] * x0.y + w[1] * x1.y + w[2] * x2.y;
  m.z = w[0] * x0.z + w[1] * x1.z + w[2] * x2.z;
  m.w = w[0] * x0.w + w[1] * x1.w + w[2] * x2.w;

  float* ap = agg + (size_t)dst * OUT_DIM + c;
  atomAddF32(ap + 0, m.x);
  atomAddF32(ap + 1, m.y);
  atomAddF32(ap + 2, m.z);
  atomAddF32(ap + 3, m.w);
  if (lane == 0) atomAddF32(deg + dst, 1.0f);
}

// ---------------------------------------------------------------------------
// Kernel 3: h = agg / max(deg,1) + bias  (stored to d_out as temp) and
// per-channel sum / sumsq partials (LDS reduce, then one atomic per channel
// per 32-node block).
// ---------------------------------------------------------------------------
__global__ __launch_bounds__(256) void bn_partial(const float* __restrict__ agg,
                                                  const float* __restrict__ deg,
                                                  const float* __restrict__ bias,
                                                  float* __restrict__ h,
                                                  float* __restrict__ bnsum,
                                                  float* __restrict__ bnsq,
                                                  int nNodes) {
  __shared__ float s_sum[256];
  __shared__ float s_sq[256];
  const int c = threadIdx.x & 127;
  const int g = threadIdx.x >> 7;    // 0 or 1
  const int nodeBase = blockIdx.x * 32;
  const float b = bias[c];

  float sum = 0.f, sq = 0.f;
  for (int i = g; i < 32; i += 2) {
    int node = nodeBase + i;
    if (node < nNodes) {
      float d = deg[node];
      d = d > 1.f ? d : 1.f;
      float v = agg[(size_t)node * OUT_DIM + c] / d + b;
      h[(size_t)node * OUT_DIM + c] = v;
      sum += v;
      sq  += v * v;
    }
  }
  s_sum[threadIdx.x] = sum;
  s_sq[threadIdx.x]  = sq;
  __syncthreads();
  if (g == 0) {
    atomAddF32(&bnsum[c], s_sum[c] + s_sum[c + 128]);
    atomAddF32(&bnsq[c],  s_sq[c]  + s_sq[c + 128]);
  }
}

// ---------------------------------------------------------------------------
// Kernel 4: fold batch statistics into per-channel scale/shift.
// ---------------------------------------------------------------------------
__global__ __launch_bounds__(128) void bn_reduce(const float* __restrict__ bnsum,
                                                 const float* __restrict__ bnsq,
                                                 const float* __restrict__ gamma,
                                                 const float* __restrict__ beta,
                                                 float* __restrict__ scale,
                                                 float* __restrict__ shift,
                                                 int nNodes) {
  int c = threadIdx.x;
  float inv  = 1.0f / (float)nNodes;
  float mean = bnsum[c] * inv;
  float var  = bnsq[c] * inv - mean * mean;
  float s    = gamma[c] * rsqrtf(var + BN_EPS);
  scale[c] = s;
  shift[c] = beta[c] - mean * s;
}

// ---------------------------------------------------------------------------
// Kernel 5: out = feature + relu(h * scale + shift), float4 streaming.
// ---------------------------------------------------------------------------
__global__ __launch_bounds__(256) void bn_final(const float* __restrict__ feature,
                                                const float* __restrict__ scale,
                                                const float* __restrict__ shift,
                                                float* __restrict__ out,
                                                int total4) {
  int i = blockIdx.x * 256 + threadIdx.x;
  if (i >= total4) return;
  int c4 = i & 31;   // 32 float4 per 128-channel row
  float4 s  = ((const float4*)scale)[c4];
  float4 sh = ((const float4*)shift)[c4];
  float4 h  = ((float4*)out)[i];
  float4 f  = ((const float4*)feature)[i];
  float4 r;
  r.x = f.x + fmaxf(h.x * s.x + sh.x, 0.f);
  r.y = f.y + fmaxf(h.y * s.y + sh.y, 0.f);
  r.z = f.z + fmaxf(h.z * s.z + sh.z, 0.f);
  r.w = f.w + fmaxf(h.w * s.w + sh.w, 0.f);
  ((float4*)out)[i] = r;
}

// ---------------------------------------------------------------------------
// Host launcher.
// ---------------------------------------------------------------------------
extern "C" void kernel_launch(void* const* d_in, const int* in_sizes, int n_in,
                              void* d_out, int out_size, void* d_ws, size_t ws_size,
                              hipStream_t stream) {
  const float* feature   = (const float*)d_in[0];
  const float* edge_attr = (const float*)d_in[1];
  const float* weight    = (const float*)d_in[2];
  const float* mu        = (const float*)d_in[3];
  const float* isig      = (const float*)d_in[4];
  const float* bias      = (const float*)d_in[5];
  const float* gamma     = (const float*)d_in[6];
  const float* beta      = (const float*)d_in[7];
  const int*   eidx      = (const int*)d_in[8];   // integer input -> const int*

  const int nNodes = in_sizes[0] / IN_DIM;
  const int nEdges = in_sizes[1] / 2;

  float* out = (float*)d_out;
  float* ws  = (float*)d_ws;

  // Workspace layout (floats):
  float* xp    = ws;                                  // N * 384
  float* agg   = xp  + (size_t)nNodes * KOUT;         // N * 128
  float* deg   = agg + (size_t)nNodes * OUT_DIM;      // N
  float* bnsum = deg + nNodes;                        // 128
  float* bnsq  = bnsum + 128;                         // 128
  float* scale = bnsq  + 128;                         // 128 (fully written, no zeroing)
  float* shift = scale + 128;                         // 128

  // 0) zero agg + deg + BN accumulators (must happen every call).
  long long nz = (long long)nNodes * OUT_DIM + nNodes + 256;
  zero_f32<<<2048, 256, 0, stream>>>(agg, nz);

  // 1) xp = feature @ weight  (WMMA f32, async-LDS double-buffered)
  dim3 gGrid(KOUT / 64, (nNodes + 63) / 64);
  gemm_xp<<<gGrid, 256, 0, stream>>>(feature, weight, xp, nNodes);

  // 2) edge aggregation (gather + fused K-blend + atomic scatter)
  edge_agg<<<(nEdges + 7) / 8, 256, 0, stream>>>(xp, edge_attr, eidx, mu, isig,
                                                 agg, deg, nEdges);

  // 3) h + BN partial sums (h parked in d_out)
  bn_partial<<<(nNodes + 31) / 32, 256, 0, stream>>>(agg, deg, bias, out,
                                                     bnsum, bnsq, nNodes);

  // 4) per-channel scale/shift
  bn_reduce<<<1, 128, 0, stream>>>(bnsum, bnsq, gamma, beta, scale, shift, nNodes);

  // 5) BN apply + ReLU + residual
  int total4 = nNodes * (OUT_DIM / 4);
  bn_final<<<(total4 + 255) / 256, 256, 0, stream>>>(feature, scale, shift, out, total4);
}